// PadWithin2D_84576495992957
// MI455X (gfx1250) — compile-verified
//
#include <hip/hip_runtime.h>

// Zero-insertion 2x upsample (stride-2 "pad within"), f32.
// in : (16*64, 128, 128) f32 = 64 MiB     out: (16*64, 256, 256) f32 = 256 MiB
//
// Zero-FLOP op -> pure HBM roofline: 320 MiB total @ 23.3 TB/s ~= 14 us.
// Mapping: one thread owns 4 grid-strided output float4's. The grid stride
// (n4/4 float4) is an exact multiple of the per-image float4 count, so the
// row phase (x4, y, even/odd) is identical across the 4 iterations: index
// math and the zero-row branch are evaluated once, then 4 coalesced NT
// b128 stores (+ 4 coalesced NT b64 loads on even rows) amortize it.

typedef __attribute__((ext_vector_type(4))) float f32x4;
typedef __attribute__((ext_vector_type(2))) float f32x2;

#define UNROLL 4

__global__ __launch_bounds__(256) void PadWithin2D_84576495992957_kernel(
    const float* __restrict__ in, float* __restrict__ out,
    unsigned n_threads,      // n4 / UNROLL; also the float4 grid stride
    unsigned in_stride)      // input float stride per iteration
{
    const unsigned idx = blockIdx.x * blockDim.x + threadIdx.x;
    if (idx >= n_threads) return;

    // Geometry (fixed by reference): Wout=256 -> 64 float4/row, Hout=256.
    const unsigned x4  = idx & 63u;   // float4 column in output row
    const unsigned row = idx >> 6;    // global output row
    const unsigned y   = row & 255u;  // output row within image
    const unsigned bc  = row >> 8;    // fused batch*channel image index

    f32x4* __restrict__ outv = (f32x4*)out;

    if (y & 1u) {
        // Odd output rows: all zeros. 4 coalesced 128-bit NT stores.
        const f32x4 z = (f32x4){0.0f, 0.0f, 0.0f, 0.0f};
#pragma unroll
        for (unsigned k = 0; k < UNROLL; ++k)
            __builtin_nontemporal_store(z, outv + idx + k * n_threads);
    } else {
        // Even rows: out[4*x4 .. 4*x4+3] = {in[2*x4], 0, in[2*x4+1], 0}.
        // Branch + offset math hoisted out of the unrolled loop; only the
        // image index advances per iteration (stride = whole images).
        const unsigned in_off = (bc * 128u + (y >> 1)) * 128u + 2u * x4;
#pragma unroll
        for (unsigned k = 0; k < UNROLL; ++k) {
            const f32x2* src = (const f32x2*)(in + in_off + k * in_stride);
            f32x2 v = __builtin_nontemporal_load(src);          // b64 NT, coalesced
            f32x4 o = (f32x4){v.x, 0.0f, v.y, 0.0f};
            __builtin_nontemporal_store(o, outv + idx + k * n_threads); // b128 NT
        }
    }
}

extern "C" void kernel_launch(void* const* d_in, const int* in_sizes, int n_in,
                              void* d_out, int out_size, void* d_ws, size_t ws_size,
                              hipStream_t stream) {
    const float* feats = (const float*)d_in[0];
    float* out = (float*)d_out;

    const unsigned n4        = (unsigned)(out_size >> 2);  // output float4 count
    const unsigned n_threads = n4 / UNROLL;                // 4,194,304
    // Per-iteration input advance in floats: (grid stride float4) spans
    // n_threads/16384 images; each image is 16384 input floats.
    const unsigned in_stride = (n_threads / 16384u) * 16384u; // = n_threads here

    const int threads = 256;                               // 8 wave32 / block
    const int blocks  = (int)((n_threads + threads - 1) / threads); // 16384

    PadWithin2D_84576495992957_kernel<<<blocks, threads, 0, stream>>>(
        feats, out, n_threads, in_stride);
}